// DeformableConv2d_7112465842416
// MI455X (gfx1250) — compile-verified
//
#include <hip/hip_runtime.h>

#define BATCH 8
#define C_IN 32
#define H 128
#define W 128
#define C_OUT 64
#define KK 9
#define KRED (C_IN * KK)   // 288
#define HW (H * W)

typedef __attribute__((ext_vector_type(2))) float v2f;
typedef __attribute__((ext_vector_type(8))) float v8f;

// ---------------------------------------------------------------------------
// Kernel 1: offset + mask 3x3 convolutions (27 output channels), zero padding.
// offY/offX clipped to +-32, mask = 2*sigmoid. One thread per pixel.
// ---------------------------------------------------------------------------
__global__ __launch_bounds__(256) void offmask_kernel(
    const float* __restrict__ x,
    const float* __restrict__ w_off, const float* __restrict__ b_off,
    const float* __restrict__ w_mod, const float* __restrict__ b_mod,
    float* __restrict__ offY, float* __restrict__ offX, float* __restrict__ Msk)
{
  // wl[(c*9+t)*28 + j]: weight for output channel j at input tap (c, t).
  // Stride 28 -> 112B rows, 16B aligned, so the j-loop vectorizes to ds_load_b128.
  __shared__ float wl[KRED * 28];
  __shared__ float bl[27];

  const int tid = threadIdx.x;
  for (int i = tid; i < KRED * 27; i += 256) {
    int ct = i / 27, j = i % 27;
    wl[ct * 28 + j] = (j < 18) ? w_off[j * KRED + ct] : w_mod[(j - 18) * KRED + ct];
  }
  if (tid < 27) bl[tid] = (tid < 18) ? b_off[tid] : b_mod[tid - 18];
  __syncthreads();

  const int p  = blockIdx.x * 256 + tid;
  const int b  = p / HW;
  const int hw = p % HW;
  const int h  = hw / W;
  const int w  = hw % W;

  float acc[27];
  #pragma unroll
  for (int j = 0; j < 27; ++j) acc[j] = 0.f;

  for (int c = 0; c < C_IN; ++c) {
    const float* xp = x + (size_t)(b * C_IN + c) * HW;
    float xv[9];
    #pragma unroll
    for (int ky = 0; ky < 3; ++ky) {
      int yy = h + ky - 1;
      bool yok = (yy >= 0) && (yy < H);
      #pragma unroll
      for (int kx = 0; kx < 3; ++kx) {
        int xx = w + kx - 1;
        bool ok = yok && (xx >= 0) && (xx < W);
        xv[ky * 3 + kx] = ok ? xp[yy * W + xx] : 0.f;
      }
    }
    #pragma unroll
    for (int t = 0; t < 9; ++t) {
      const float v = xv[t];
      const float* wp = &wl[(c * 9 + t) * 28];
      #pragma unroll
      for (int j = 0; j < 27; ++j) acc[j] = fmaf(v, wp[j], acc[j]);
    }
  }

  const float max_off = 32.f;  // max(H, W) / 4
  #pragma unroll
  for (int t = 0; t < 9; ++t) {
    float oy = acc[2 * t]     + bl[2 * t];
    float ox = acc[2 * t + 1] + bl[2 * t + 1];
    oy = fminf(fmaxf(oy, -max_off), max_off);
    ox = fminf(fmaxf(ox, -max_off), max_off);
    float z = acc[18 + t] + bl[18 + t];
    float m = 2.f / (1.f + __expf(-z));
    int idx = ((b * 9 + t) * H + h) * W + w;
    offY[idx] = oy;
    offX[idx] = ox;
    Msk[idx]  = m;
  }
}

// ---------------------------------------------------------------------------
// Kernel 2: bilinear gather into LDS B-tile, then f32 WMMA GEMM:
//   out(64 x 32pix) = w_reg(64 x 288) * sampled(288 x 32pix)
// Block = 256 threads (8 wave32): 4 out-channel tiles x 2 pixel tiles.
//
// LDS strides chosen so (2*stride) mod 64 == 16: the two half-waves of a
// WMMA fragment read (rows k/k+1 vs k+2/k+3) hit DISJOINT 16-bank groups
// -> conflict-free ds_load_b32 for both A and B fragments.
// ---------------------------------------------------------------------------
#define WLDS_STRIDE 72                      // 2*72 = 144 = 16 (mod 64)
#define BT_STRIDE   40                      // 2*40 =  80 = 16 (mod 64)
#define SMEM2_FLOATS (KRED * WLDS_STRIDE + KRED * BT_STRIDE)  // 32256 fl = 126 KB

__global__ __launch_bounds__(256) void dconv_wmma_kernel(
    const float* __restrict__ x, const float* __restrict__ w_reg,
    const float* __restrict__ offY, const float* __restrict__ offX,
    const float* __restrict__ Msk, float* __restrict__ out)
{
  extern __shared__ float smem[];
  float* Wlds  = smem;                       // [k][o], stride 72
  float* Btile = smem + KRED * WLDS_STRIDE;  // [k][pix], stride 40

  const int tid = threadIdx.x;
  const int seg = blockIdx.x & 3;            // 32-pixel segment in row
  const int h   = (blockIdx.x >> 2) & (H - 1);
  const int b   = blockIdx.x >> 9;

  // Stage w_reg transposed: Wlds[k][o] = w_reg[o*288 + k]. Coalesced global
  // reads; (o,k) tracked incrementally to avoid per-iteration div/mod.
  {
    int k = tid & 255;                       // tid < 256 < KRED, so o starts 0
    int o = 0;
    #pragma unroll
    for (int it = 0; it < (C_OUT * KRED) / 256; ++it) {
      Wlds[k * WLDS_STRIDE + o] = w_reg[o * KRED + k];
      k += 256;
      if (k >= KRED) { k -= KRED; ++o; }
    }
  }

  // Build B tile: Btile[c*9+t][pix] = mask * bilinear(x, py, px).
  const int pix = tid & 31;
  const int cg  = tid >> 5;                  // 0..7 -> channels {cg, cg+8, cg+16, cg+24}
  const int wo  = seg * 32 + pix;

  #pragma unroll
  for (int t = 0; t < 9; ++t) {
    const int om = ((b * 9 + t) * H + h) * W + wo;
    const float oy = offY[om];
    const float ox = offX[om];
    const float m  = Msk[om];
    const float py = oy + (float)(h - 1 + (t / 3));
    const float px = ox + (float)(wo - 1 + (t % 3));
    const float y0f = floorf(py), x0f = floorf(px);
    const float dy = py - y0f,    dx = px - x0f;
    const int y0 = (int)y0f, x0 = (int)x0f;
    const int y1 = y0 + 1,   x1 = x0 + 1;
    const bool yv0 = (y0 >= 0) && (y0 < H);
    const bool yv1 = (y1 >= 0) && (y1 < H);
    const bool xv0 = (x0 >= 0) && (x0 < W);
    const bool xv1 = (x1 >= 0) && (x1 < W);
    const int yc0 = min(max(y0, 0), H - 1);
    const int yc1 = min(max(y1, 0), H - 1);
    const int xc0 = min(max(x0, 0), W - 1);
    const int xc1 = min(max(x1, 0), W - 1);
    const int i00 = yc0 * W + xc0, i01 = yc0 * W + xc1;
    const int i10 = yc1 * W + xc0, i11 = yc1 * W + xc1;
    const float w00 = (yv0 && xv0) ? (1.f - dy) * (1.f - dx) * m : 0.f;
    const float w01 = (yv0 && xv1) ? (1.f - dy) * dx * m         : 0.f;
    const float w10 = (yv1 && xv0) ? dy * (1.f - dx) * m         : 0.f;
    const float w11 = (yv1 && xv1) ? dy * dx * m                 : 0.f;
    #pragma unroll
    for (int cc = 0; cc < 4; ++cc) {
      const int c = cg + cc * 8;
      const float* xp = x + (size_t)(b * C_IN + c) * HW;
      float s = w00 * xp[i00] + w01 * xp[i01] + w10 * xp[i10] + w11 * xp[i11];
      Btile[(c * 9 + t) * BT_STRIDE + pix] = s;
    }
  }
  __syncthreads();

  // WMMA stage: per wave, one 16(out-ch) x 16(pix) tile, K = 288 in steps of 4.
  // A 16x4 f32 layout: lane l<16 holds M=l, K={0,1}; lane>=16 holds M=l-16, K={2,3}.
  // B 4x16 f32 layout: VGPR v, lane<16 -> row K=v, col=lane; lane>=16 -> K=v+2.
  const int lane  = tid & 31;
  const int wave  = tid >> 5;
  const int ctile = wave & 3;          // out-channel tile (0..3)
  const int ptile = wave >> 2;         // pixel tile (0..1)
  const int l     = lane & 15;
  const int half  = lane >> 4;
  const int koff  = half * 2;

  const float* Wb = Wlds  + ctile * 16 + l;   // + k * WLDS_STRIDE
  const float* Bb = Btile + ptile * 16 + l;   // + k * BT_STRIDE

  v8f acc = {0.f, 0.f, 0.f, 0.f, 0.f, 0.f, 0.f, 0.f};
  #pragma unroll 8
  for (int k0 = 0; k0 < KRED; k0 += 4) {
    v2f a, bv;
    a[0]  = Wb[(k0 + koff)     * WLDS_STRIDE];
    a[1]  = Wb[(k0 + koff + 1) * WLDS_STRIDE];
    bv[0] = Bb[(k0 + koff)     * BT_STRIDE];
    bv[1] = Bb[(k0 + koff + 1) * BT_STRIDE];
    acc = __builtin_amdgcn_wmma_f32_16x16x4_f32(false, a, false, bv,
                                                (short)0, acc, false, false);
  }

  // C/D layout: VGPR r -> M = r + 8*half, N = l.
  const int wo2 = seg * 32 + ptile * 16 + l;
  const int chb = ctile * 16 + half * 8;
  #pragma unroll
  for (int r = 0; r < 8; ++r) {
    out[(((size_t)(b * C_OUT + chb + r)) * H + h) * W + wo2] = acc[r];
  }
}

// ---------------------------------------------------------------------------
extern "C" void kernel_launch(void* const* d_in, const int* in_sizes, int n_in,
                              void* d_out, int out_size, void* d_ws, size_t ws_size,
                              hipStream_t stream) {
  const float* x     = (const float*)d_in[0];
  const float* w_off = (const float*)d_in[1];
  const float* b_off = (const float*)d_in[2];
  const float* w_mod = (const float*)d_in[3];
  const float* b_mod = (const float*)d_in[4];
  const float* w_reg = (const float*)d_in[5];
  float* out = (float*)d_out;

  // Workspace layout: offY | offX | mask, each BATCH*9*H*W floats (~14.2 MB total).
  const size_t plane = (size_t)BATCH * 9 * HW;
  float* offY = (float*)d_ws;
  float* offX = offY + plane;
  float* Msk  = offX + plane;

  offmask_kernel<<<(BATCH * HW) / 256, 256, 0, stream>>>(
      x, w_off, b_off, w_mod, b_mod, offY, offX, Msk);

  dconv_wmma_kernel<<<BATCH * H * (W / 32), 256,
                      SMEM2_FLOATS * sizeof(float), stream>>>(
      x, w_reg, offY, offX, Msk, out);
}